// SelectiveMambaGramLayer_32581621907456
// MI455X (gfx1250) — compile-verified
//
#include <hip/hip_runtime.h>
#include <math.h>

// ---------------------------------------------------------------------------
// SelectiveMambaGram scan for MI455X (gfx1250), 3-pass chunked formulation.
//   B=16, L=16384, D=64, HID=16.  Output: (B,L,D) complex64 = 134 MB (dominant).
// Roofline: output stream 134 MB / 23.3 TB/s ~= 5.8 us floor; all intermediates
// (4 MiB) stay L2-resident; a_t recomputed per pass (50M transcendentals beats
// 2x268 MB of HBM spill by ~20x).
// CDNA5 paths used:
//   - V_WMMA_F32_16X16X4_F32 x4 for the HID=16 MLP contraction (matrix pipe
//     co-executes with the VALU transcendentals).
//   - global_load_async_to_lds_b32 + s_wait_asynccnt for chunk staging.
// Workspace layout (needs 4 MiB):
//   [0       , 1 MiB)  delta[b,l]            (B*L floats)
//   [1 MiB   , 2 MiB)  chunkA[b,c,d] float2  (per-chunk product of a)
//   [2 MiB   , 3 MiB)  chunkH[b,c,d] float2  (per-chunk local scan end, h0=0)
//   [3 MiB   , 4 MiB)  hin[b,c,d]    float2  (incoming state per chunk)
// ---------------------------------------------------------------------------

#define B_N   16
#define L_N   16384
#define D_N   64
#define HID_N 16
#define T_CH  128             // chunk length
#define C_CH  (L_N / T_CH)    // 128 chunks per sequence

typedef __attribute__((ext_vector_type(2))) float v2f;
typedef __attribute__((ext_vector_type(8))) float v8f;

#if __has_builtin(__builtin_amdgcn_wmma_f32_16x16x4_f32)
#define HAVE_WMMA_F32X4 1
#else
#define HAVE_WMMA_F32X4 0
#endif

#if __has_builtin(__builtin_amdgcn_global_load_async_to_lds_b32)
#define HAVE_ASYNC_LDS 1
#else
#define HAVE_ASYNC_LDS 0
#endif

typedef __attribute__((address_space(1))) int g_i32;   // global int
typedef __attribute__((address_space(3))) int l_i32;   // LDS int

// v_rcp_f32 (1 ulp) instead of the IEEE div expansion — consistent with __expf.
__device__ __forceinline__ float rcp_f(float z) {
#if __has_builtin(__builtin_amdgcn_rcpf)
  return __builtin_amdgcn_rcpf(z);
#else
  return 1.0f / z;
#endif
}
__device__ __forceinline__ float sigmoid_f(float z) { return rcp_f(1.0f + __expf(-z)); }
__device__ __forceinline__ float silu_f(float z)    { return z * sigmoid_f(z); }
__device__ __forceinline__ float softplus_f(float z){ return (z > 20.0f) ? z : log1pf(__expf(z)); }

// ---------------------------------------------------------------------------
// Kernel 1: delta[b,l] via the 1->16->1 SiLU MLP.
// Each wave handles 16 timesteps; the HID=16 contraction is done on the matrix
// pipe with 4 accumulating V_WMMA_F32_16X16X4_F32 ops.
//   A[m,k]  = silu(x[t=base+m]*w1[k]+b1[k])   (16x4 f32 layout: lanes 0-15 carry
//             K=0,1 in vgpr0/1; lanes 16-31 carry K=2,3)
//   B[k,n]  = w2[k] broadcast over n          (same half-wave K split)
//   D[m,n]  = d_raw[base+m] for every n; D vgpr r = m=r (low half) / m=8+r (high)
// ---------------------------------------------------------------------------
__global__ __launch_bounds__(256) void delta_mlp_kernel(
    const float* __restrict__ x,  const float* __restrict__ w1,
    const float* __restrict__ b1, const float* __restrict__ w2,
    const float* __restrict__ b2, float* __restrict__ delta_out)
{
  const int lane = threadIdx.x & 31;
  const int wave = threadIdx.x >> 5;
  const int half = lane & 15;
  const int hi   = lane >> 4;
  const int base = (blockIdx.x * 8 + wave) * 16;
  const float b2v = b2[0];
  const float xv  = x[base + half];

#if HAVE_WMMA_F32X4
  v8f acc = {0.f, 0.f, 0.f, 0.f, 0.f, 0.f, 0.f, 0.f};
  #pragma unroll
  for (int kb = 0; kb < 4; ++kb) {
    const int k0 = kb * 4 + hi * 2;
    v2f a, bm;
    a.x  = silu_f(fmaf(xv, w1[k0],     b1[k0]));
    a.y  = silu_f(fmaf(xv, w1[k0 + 1], b1[k0 + 1]));
    bm.x = w2[k0];
    bm.y = w2[k0 + 1];
    acc = __builtin_amdgcn_wmma_f32_16x16x4_f32(false, a, false, bm,
                                                (short)0, acc, false, false);
  }
  if (half == 0) {   // lane 0 owns m=0..7, lane 16 owns m=8..15
    #pragma unroll
    for (int r = 0; r < 8; ++r) {
      const float draw = acc[r] + b2v;
      delta_out[base + hi * 8 + r] = 0.1f + 9.9f * sigmoid_f(draw);
    }
  }
#else
  float acc = 0.0f;
  #pragma unroll
  for (int k = 0; k < HID_N; ++k)
    acc = fmaf(silu_f(fmaf(xv, w1[k], b1[k])), w2[k], acc);
  if (hi == 0)
    delta_out[base + half] = 0.1f + 9.9f * sigmoid_f(acc + b2v);
#endif
}

// ---------------------------------------------------------------------------
// Stage one chunk of x and delta into LDS (broadcast-read afterwards by all 64
// channel-threads). Uses the CDNA5 async global->LDS DMA path.
// ---------------------------------------------------------------------------
__device__ __forceinline__ void stage_chunk(const float* __restrict__ gx,
                                            const float* __restrict__ gd,
                                            float* lx, float* ld)
{
#if HAVE_ASYNC_LDS
  for (int i = threadIdx.x; i < T_CH; i += 64) {
    __builtin_amdgcn_global_load_async_to_lds_b32(
        (g_i32*)(gx + i), (l_i32*)(lx + i), 0, 0);
    __builtin_amdgcn_global_load_async_to_lds_b32(
        (g_i32*)(gd + i), (l_i32*)(ld + i), 0, 0);
  }
#if __has_builtin(__builtin_amdgcn_s_wait_asynccnt)
  __builtin_amdgcn_s_wait_asynccnt(0);
#else
  asm volatile("s_wait_asynccnt 0x0" ::: "memory");
#endif
#else
  for (int i = threadIdx.x; i < T_CH; i += 64) { lx[i] = gx[i]; ld[i] = gd[i]; }
#endif
  __syncthreads();
}

// ---------------------------------------------------------------------------
// Pass 1: per (b,chunk) compute, per channel d, the chunk summary:
//   chunkA = prod_t a_t   and   chunkH = local scan end with h0 = 0.
// Thread = channel; T_CH sequential steps; a_t recomputed (cheaper than HBM).
// ---------------------------------------------------------------------------
__global__ __launch_bounds__(64) void chunk_scan_pass1(
    const float* __restrict__ x, const float* __restrict__ delta,
    const float* __restrict__ omega, const float* __restrict__ raw_alpha,
    const float* __restrict__ b_real, const float* __restrict__ b_imag,
    float2* __restrict__ chunkA, float2* __restrict__ chunkH)
{
  __shared__ float sx[T_CH];
  __shared__ float sd[T_CH];
  const int d = threadIdx.x;
  const int b = blockIdx.x / C_CH;
  const int c = blockIdx.x % C_CH;
  const size_t p0 = (size_t)b * L_N + (size_t)c * T_CH;
  stage_chunk(x + p0, delta + p0, sx, sd);

  const float al = -softplus_f(raw_alpha[d]);
  const float om = omega[d];
  const float br = b_real[d], bi = b_imag[d];
  float hr = 0.f, hi = 0.f, par = 1.f, pai = 0.f;
  #pragma unroll 4
  for (int t = 0; t < T_CH; ++t) {
    const float dt = sd[t], xv = sx[t];
    const float e = __expf(dt * al);
    float sn, cs; __sincosf(dt * om, &sn, &cs);
    const float ar = e * cs, ai = e * sn;
    const float ur = dt * xv * br, ui = dt * xv * bi;
    const float nhr = ar * hr - ai * hi + ur;
    const float nhi = ar * hi + ai * hr + ui;
    hr = nhr; hi = nhi;
    const float npr = ar * par - ai * pai;
    const float npi = ar * pai + ai * par;
    par = npr; pai = npi;
  }
  const int rec = (b * C_CH + c) * D_N + d;
  chunkA[rec] = make_float2(par, pai);
  chunkH[rec] = make_float2(hr, hi);
}

// ---------------------------------------------------------------------------
// Pass 2: scan across chunk summaries. One thread per (b,d); C_CH=128 steps.
// hin[b,c,d] = state entering chunk c.
// ---------------------------------------------------------------------------
__global__ __launch_bounds__(256) void chunk_scan_pass2(
    const float2* __restrict__ chunkA, const float2* __restrict__ chunkH,
    float2* __restrict__ hin)
{
  const int idx = blockIdx.x * blockDim.x + threadIdx.x;
  if (idx >= B_N * D_N) return;
  const int b = idx / D_N;
  const int d = idx % D_N;
  float sr = 0.f, si = 0.f;
  for (int c = 0; c < C_CH; ++c) {
    const int rec = (b * C_CH + c) * D_N + d;
    hin[rec] = make_float2(sr, si);
    const float2 A = chunkA[rec];
    const float2 H = chunkH[rec];
    const float nr = A.x * sr - A.y * si + H.x;
    const float ni = A.x * si + A.y * sr + H.y;
    sr = nr; si = ni;
  }
}

// ---------------------------------------------------------------------------
// Pass 3: recompute the chunk-local scan seeded with hin and stream the output.
// At each t the 64 channel-threads store one contiguous 512 B float2 segment.
// ---------------------------------------------------------------------------
__global__ __launch_bounds__(64) void chunk_scan_pass3(
    const float* __restrict__ x, const float* __restrict__ delta,
    const float* __restrict__ omega, const float* __restrict__ raw_alpha,
    const float* __restrict__ b_real, const float* __restrict__ b_imag,
    const float2* __restrict__ hin, float2* __restrict__ out)
{
  __shared__ float sx[T_CH];
  __shared__ float sd[T_CH];
  const int d = threadIdx.x;
  const int b = blockIdx.x / C_CH;
  const int c = blockIdx.x % C_CH;
  const size_t p0 = (size_t)b * L_N + (size_t)c * T_CH;
  stage_chunk(x + p0, delta + p0, sx, sd);

  const float al = -softplus_f(raw_alpha[d]);
  const float om = omega[d];
  const float br = b_real[d], bi = b_imag[d];
  const float2 h0 = hin[(b * C_CH + c) * D_N + d];
  float hr = h0.x, hi = h0.y;
  #pragma unroll 4
  for (int t = 0; t < T_CH; ++t) {
    const float dt = sd[t], xv = sx[t];
    const float e = __expf(dt * al);
    float sn, cs; __sincosf(dt * om, &sn, &cs);
    const float ar = e * cs, ai = e * sn;
    const float ur = dt * xv * br, ui = dt * xv * bi;
    const float nhr = ar * hr - ai * hi + ur;
    const float nhi = ar * hi + ai * hr + ui;
    hr = nhr; hi = nhi;
    out[(p0 + (size_t)t) * D_N + d] = make_float2(hr, hi);
  }
}

// ---------------------------------------------------------------------------
extern "C" void kernel_launch(void* const* d_in, const int* in_sizes, int n_in,
                              void* d_out, int out_size, void* d_ws, size_t ws_size,
                              hipStream_t stream) {
  (void)in_sizes; (void)n_in; (void)out_size; (void)ws_size;
  const float* x         = (const float*)d_in[0];
  const float* omega     = (const float*)d_in[1];
  const float* raw_alpha = (const float*)d_in[2];
  const float* b_real    = (const float*)d_in[3];
  const float* b_imag    = (const float*)d_in[4];
  const float* w1        = (const float*)d_in[5];
  const float* b1        = (const float*)d_in[6];
  const float* w2        = (const float*)d_in[7];
  const float* b2        = (const float*)d_in[8];

  char* ws = (char*)d_ws;                      // needs 4 MiB total
  float*  delta  = (float*)(ws);               // B*L floats           (1 MiB)
  float2* chunkA = (float2*)(ws + (1u << 20)); // B*C*D float2         (1 MiB)
  float2* chunkH = (float2*)(ws + (2u << 20)); // B*C*D float2         (1 MiB)
  float2* hin    = (float2*)(ws + (3u << 20)); // B*C*D float2         (1 MiB)
  float2* out    = (float2*)d_out;             // (B,L,D) complex64 interleaved

  // 1) delta MLP on the WMMA pipe: (B*L)/(8 waves * 16 t) = 2048 blocks.
  delta_mlp_kernel<<<(B_N * L_N) / (8 * 16), 256, 0, stream>>>(x, w1, b1, w2, b2, delta);
  // 2) per-chunk summaries: B*C = 2048 blocks, 64 channel-threads each.
  chunk_scan_pass1<<<B_N * C_CH, 64, 0, stream>>>(x, delta, omega, raw_alpha,
                                                  b_real, b_imag, chunkA, chunkH);
  // 3) inter-chunk scan: B*D = 1024 sequences of C=128 records.
  chunk_scan_pass2<<<(B_N * D_N + 255) / 256, 256, 0, stream>>>(chunkA, chunkH, hin);
  // 4) final scan + coalesced 134 MB output stream (the bandwidth floor).
  chunk_scan_pass3<<<B_N * C_CH, 64, 0, stream>>>(x, delta, omega, raw_alpha,
                                                  b_real, b_imag, hin, out);
}